// CNN_align_85564338471617
// MI455X (gfx1250) — compile-verified
//
#include <hip/hip_runtime.h>
#include <hip/hip_bf16.h>
#include <stdint.h>

typedef float        f32x2 __attribute__((ext_vector_type(2)));
typedef float        f32x4 __attribute__((ext_vector_type(4)));
typedef float        f32x8 __attribute__((ext_vector_type(8)));
typedef unsigned int u32x4 __attribute__((ext_vector_type(4)));
typedef int          i32x4 __attribute__((ext_vector_type(4)));
typedef int          i32x8 __attribute__((ext_vector_type(8)));

#define BATCH 8
#define HH 48
#define WW 48
#define IJ (HH * WW)        // 2304 inner (i,j) elements per (b,y,x)
#define K2_THREADS 192      // 6 wave32 waves; 576 float4 / 192 = 3 segments/thread

// ---------------------------------------------------------------------------
// Kernel 1: per-batch TPS fit (12x12 Gauss-Jordan, 2 RHS) + grid evaluation.
// Writes gxy[b][0][ij] = gx, gxy[b][1][ij] = gy, and zeroes the output sums.
// ---------------------------------------------------------------------------
__global__ __launch_bounds__(256) void tps_grid_kernel(
    const float* __restrict__ mv,    // [8][9][2]
    float* __restrict__ gxy,         // [8][2][2304]
    float* __restrict__ sums)        // [8]
{
    const int b = blockIdx.x;
    __shared__ float s_th[2][12];    // theta_x, theta_y (full 12-vector)
    __shared__ float s_dst[9][2];

    if (threadIdx.x == 0) {
        sums[b] = 0.0f;              // output sums must start at zero each call
        const float src[9][2] = {{0.f,0.f},{0.5f,0.f},{1.f,0.f},
                                 {0.f,0.5f},{0.5f,0.5f},{1.f,0.5f},
                                 {0.f,1.f},{5.f,1.f},{1.f,1.f}};
        float dst[9][2];
        for (int k = 0; k < 9; ++k) {
            dst[k][0] = src[k][0] + mv[(b*9 + k)*2 + 0];
            dst[k][1] = src[k][1] + mv[(b*9 + k)*2 + 1];
            s_dst[k][0] = dst[k][0];
            s_dst[k][1] = dst[k][1];
        }
        // Augmented system [A | rhs_x rhs_y], A is the standard TPS matrix.
        float A[12][14];
        for (int i = 0; i < 12; ++i)
            for (int j = 0; j < 14; ++j) A[i][j] = 0.0f;
        for (int i = 0; i < 9; ++i)
            for (int j = 0; j < 9; ++j) {
                float ddx = dst[i][0] - dst[j][0];
                float ddy = dst[i][1] - dst[j][1];
                float r   = sqrtf(ddx*ddx + ddy*ddy);
                A[i][j]   = r * r * logf(r + 1e-6f);
            }
        for (int i = 0; i < 9; ++i) {
            A[i][9]  = 1.0f; A[i][10] = dst[i][0]; A[i][11] = dst[i][1];
            A[9][i]  = 1.0f; A[10][i] = dst[i][0]; A[11][i] = dst[i][1];
            A[i][12] = -mv[(b*9 + i)*2 + 0];   // delta = src - dst = -mv
            A[i][13] = -mv[(b*9 + i)*2 + 1];
        }
        // Gauss-Jordan with partial pivoting (zero diagonal block requires it)
        for (int col = 0; col < 12; ++col) {
            int piv = col; float best = fabsf(A[col][col]);
            for (int r2 = col + 1; r2 < 12; ++r2) {
                float v = fabsf(A[r2][col]);
                if (v > best) { best = v; piv = r2; }
            }
            if (piv != col)
                for (int j = 0; j < 14; ++j) {
                    float t = A[col][j]; A[col][j] = A[piv][j]; A[piv][j] = t;
                }
            float inv = 1.0f / A[col][col];
            for (int j = 0; j < 14; ++j) A[col][j] *= inv;
            for (int r2 = 0; r2 < 12; ++r2) {
                if (r2 == col) continue;
                float f = A[r2][col];
                for (int j = 0; j < 14; ++j) A[r2][j] -= f * A[col][j];
            }
        }
        // theta[0..8]=weights (theta[0] == -sum(theta[1..8]) by constraint),
        // theta[9..11]=affine -> matches reference's reconstructed w exactly.
        for (int i = 0; i < 12; ++i) { s_th[0][i] = A[i][12]; s_th[1][i] = A[i][13]; }
    }
    __syncthreads();

    for (int p = threadIdx.x; p < IJ; p += blockDim.x) {
        int i = p / WW, j = p % WW;
        float x = (float)j * (1.0f / 47.0f);
        float y = (float)i * (1.0f / 47.0f);
        float zx = s_th[0][9] + s_th[0][10]*x + s_th[0][11]*y;
        float zy = s_th[1][9] + s_th[1][10]*x + s_th[1][11]*y;
        #pragma unroll
        for (int k = 0; k < 9; ++k) {
            float ddx = x - s_dst[k][0];
            float ddy = y - s_dst[k][1];
            float r   = sqrtf(ddx*ddx + ddy*ddy);
            float u   = r * r * logf(r + 1e-6f);
            zx += s_th[0][k] * u;
            zy += s_th[1][k] * u;
        }
        gxy[(size_t)b*2*IJ + p]      = (x + zx) * 47.0f;   // gx
        gxy[(size_t)b*2*IJ + IJ + p] = (y + zy) * 47.0f;   // gy
    }
}

// ---------------------------------------------------------------------------
// Kernel 2: one block per (b, y). TDM DMA of the batch gx/gy table into LDS,
// then stream 48 x-slabs (2304 f32 each) with b128 NT loads/stores, masking
// and accumulating the per-batch sum. Per-wave reduction via WMMA f32 16x16x4
// against an all-ones B matrix (row-sum trick).
// ---------------------------------------------------------------------------
__global__ __launch_bounds__(K2_THREADS) void mask_sum_kernel(
    const float* __restrict__ corr,  // [8][48][48][48][48]
    const float* __restrict__ gxy,   // [8][2][2304]
    float* __restrict__ out,         // [8][48][48][48][48]
    float* __restrict__ sums)        // [8]
{
    __shared__ __align__(16) float sh[2 * IJ];   // gx | gy for this batch
    __shared__ float red[K2_THREADS / 32];       // one partial per wave

    const int b = blockIdx.x / HH;
    const int y = blockIdx.x % HH;
    const int t = threadIdx.x;

    // --- CDNA5 Tensor Data Mover: 1-D tile (4608 f32) global -> LDS -------
    // Only wave 0 issues the DMA (scalar-uniform branch); TENSORcnt is
    // per-wave, so wave 0 waits, then the workgroup barrier publishes LDS.
    if (__builtin_amdgcn_readfirstlane((int)t) == 0) {
        uint64_t ga  = (uint64_t)(uintptr_t)(gxy + (size_t)b * 2 * IJ);
        uint32_t lds = (uint32_t)(uintptr_t)&sh[0];  // flat->LDS: low 32 bits

        u32x4 g0; i32x8 g1; i32x4 g2, g3; i32x8 g4;
        // D# group0: count=1 (valid), lds_addr, 57b global_addr, type=2
        g0[0] = 1u;
        g0[1] = lds;
        g0[2] = (uint32_t)(ga & 0xFFFFFFFFu);
        g0[3] = (uint32_t)((ga >> 32) & 0x01FFFFFFu) | 0x80000000u;
        // D# group1: mask=0, data_size=2 (4B); tensor_dim0=4608; dim1=1;
        // tile_dim0=4608; tile_dim1=1; stride0=stride1=4608
        g1[0] = (int)(2u << 16);
        g1[1] = (int)((4608u & 0xFFFFu) << 16); // dim0[15:0] @ bits 63:48
        g1[2] = (int)(1u << 16);                // dim0 hi=0 | dim1 lo=1
        g1[3] = (int)(4608u << 16);             // dim1 hi=0 | tile_dim0=4608
        g1[4] = 1;                              // tile_dim1=1 | tile_dim2=0
        g1[5] = 4608;                           // stride0 lo
        g1[6] = (int)(4608u << 16);             // stride0 hi=0 | stride1 lo
        g1[7] = 0;                              // stride1 hi
        // D# group2/3: unused dims = 1 (avoid OOB-zeroing), tile dims = 0
        g2[0] = 1;    g2[1] = 1;               g2[2] = 4608; g2[3] = 0;
        g3[0] = 4608; g3[1] = (int)(1u << 16); g3[2] = 0;    g3[3] = 0;
        // Extra descriptor group (6-arg toolchain form): zero-filled
        g4[0] = 0; g4[1] = 0; g4[2] = 0; g4[3] = 0;
        g4[4] = 0; g4[5] = 0; g4[6] = 0; g4[7] = 0;

        __builtin_amdgcn_tensor_load_to_lds(g0, g1, g2, g3, g4, 0);
        __builtin_amdgcn_s_wait_tensorcnt((short)0);
    }
    __syncthreads();

    // Register-resident gx/gy + precomputed y-mask (fixed per block)
    f32x4 gx4[3], gy4[3], cym[3];
    const float fy = (float)y;
    #pragma unroll
    for (int s = 0; s < 3; ++s) {
        const int q = t + s * K2_THREADS;           // float4 index in [0,576)
        gx4[s] = *(const f32x4*)&sh[q * 4];
        gy4[s] = *(const f32x4*)&sh[IJ + q * 4];
        cym[s].x = (fabsf(fy - gy4[s].x) <= 1.0f) ? 1.0f : 0.0f;
        cym[s].y = (fabsf(fy - gy4[s].y) <= 1.0f) ? 1.0f : 0.0f;
        cym[s].z = (fabsf(fy - gy4[s].z) <= 1.0f) ? 1.0f : 0.0f;
        cym[s].w = (fabsf(fy - gy4[s].w) <= 1.0f) ? 1.0f : 0.0f;
    }

    const size_t base = ((size_t)b * HH + y) * WW * IJ;
    const float* cptr = corr + base;
    float*       optr = out  + base;

    float acc = 0.0f;
    for (int x = 0; x < WW; ++x) {
        const float  fx = (float)x;
        const size_t xb = (size_t)x * IJ;
        #pragma unroll
        for (int s = 0; s < 3; ++s) {
            const int off = (t + s * K2_THREADS) * 4;
            f32x4 c = __builtin_nontemporal_load((const f32x4*)(cptr + xb + off));
            f32x4 m;
            m.x = ((fabsf(fx - gx4[s].x) <= 1.0f) ? cym[s].x : 0.0f) * c.x;
            m.y = ((fabsf(fx - gx4[s].y) <= 1.0f) ? cym[s].y : 0.0f) * c.y;
            m.z = ((fabsf(fx - gx4[s].z) <= 1.0f) ? cym[s].z : 0.0f) * c.z;
            m.w = ((fabsf(fx - gx4[s].w) <= 1.0f) ? cym[s].w : 0.0f) * c.w;
            __builtin_nontemporal_store(m, (f32x4*)(optr + xb + off));
            acc += (m.x + m.y) + (m.z + m.w);
        }
    }

    // --- Per-wave reduction via V_WMMA_F32_16X16X4_F32 --------------------
    // A (16x4): lanes 0-15 hold {K0=acc, K1=0}, lanes 16-31 {K2=acc, K3=0}.
    // B = all-ones 4x16  ->  D[m,n] = acc_m + acc_{m+16} for every column n.
    // Lane n<16 holds D rows 0..7 in 8 VGPRs; lane n+16 holds rows 8..15.
    // All 192 threads are converged here (EXEC all ones, WMMA requirement).
    f32x2 a;  a.x = acc;  a.y = 0.0f;
    f32x2 bo; bo.x = 1.0f; bo.y = 1.0f;
    f32x8 dmat = {};
    dmat = __builtin_amdgcn_wmma_f32_16x16x4_f32(
        false, a, false, bo, (short)0, dmat, false, false);
    float s8 = ((dmat[0] + dmat[1]) + (dmat[2] + dmat[3])) +
               ((dmat[4] + dmat[5]) + (dmat[6] + dmat[7]));
    // add partner half (rows 8..15 live in lane^16); identical per column
    float wsum = s8 + __shfl(s8, (t & 31) ^ 16, 32);

    if ((t & 31) == 0) red[t >> 5] = wsum;
    __syncthreads();
    if (t == 0) {
        float bsum = ((red[0] + red[1]) + (red[2] + red[3])) + (red[4] + red[5]);
        atomicAdd(sums + b, bsum);
    }
}

// ---------------------------------------------------------------------------
extern "C" void kernel_launch(void* const* d_in, const int* in_sizes, int n_in,
                              void* d_out, int out_size, void* d_ws, size_t ws_size,
                              hipStream_t stream) {
    const float* mv   = (const float*)d_in[0];   // geo_parameters [8,9,2]
    const float* corr = (const float*)d_in[1];   // corr_scores [8,48,48,48,48]
    float* out  = (float*)d_out;
    float* sums = out + (size_t)BATCH * HH * WW * IJ;   // after 8*48^4 floats
    float* gxy  = (float*)d_ws;                         // 8*4608 f32 = 147 KB

    tps_grid_kernel<<<BATCH, 256, 0, stream>>>(mv, gxy, sums);
    mask_sum_kernel<<<BATCH * HH, K2_THREADS, 0, stream>>>(corr, gxy, out, sums);
}